// SampleAndAggregate_64295660421648
// MI455X (gfx1250) — compile-verified
//
#include <hip/hip_runtime.h>

// ---- problem constants (match reference) ----
#define FEAT   128      // feature dim == DIMS[0] == DIMS[1] == DIMS[2]
#define MAXDEG 128
#define BATCH  512
#define S1     10       // NUM_SAMPLES[1]  (hop-1 fanout)
#define S0     25       // NUM_SAMPLES[0]  (hop-2 fanout)
#define N1     (BATCH * S1)   // 5120
#define OUTD   256      // concat width (2 * 128)

typedef __attribute__((ext_vector_type(2))) float v2f;
typedef __attribute__((ext_vector_type(8))) float v8f;

// n1[i] = adj[batch1[i/10] * MAXDEG + cols1[i]]
__global__ void k_compute_n1(const int* __restrict__ adj,
                             const int* __restrict__ batch1,
                             const int* __restrict__ cols1,
                             int* __restrict__ n1) {
  int i = blockIdx.x * blockDim.x + threadIdx.x;
  if (i < N1) {
    int b = batch1[i / S1];
    int c = cols1[i];
    n1[i] = adj[(long long)b * MAXDEG + c];
  }
}

// dst[r, :] = feat[idx[r], :]   (one block per row, 128 threads, coalesced)
__global__ void k_gather(const float* __restrict__ feat,
                         const int* __restrict__ idx,
                         float* __restrict__ dst) {
  int r = blockIdx.x;
  int t = threadIdx.x;
  dst[(long long)r * FEAT + t] = feat[(long long)idx[r] * FEAT + t];
}

// out[r, :] = mean_k feat[ adj[n1[r]*MAXDEG + cols2[r*25+k]], : ]   (the 65 MB gather)
__global__ void k_mean25(const float* __restrict__ feat,
                         const int* __restrict__ adj,
                         const int* __restrict__ n1,
                         const int* __restrict__ cols2,
                         float* __restrict__ out) {
  int r = blockIdx.x;
  int t = threadIdx.x;
  long long abase = (long long)n1[r] * MAXDEG;
  float acc = 0.f;
  #pragma unroll 5
  for (int k = 0; k < S0; ++k) {
    int nb = adj[abase + cols2[r * S0 + k]];
    acc += feat[(long long)nb * FEAT + t];
  }
  out[(long long)r * FEAT + t] = acc * (1.0f / S0);
}

// dst[r, :] = mean over NS consecutive rows of src (row width DIM)
template <int DIM, int NS>
__global__ void k_mean_rows(const float* __restrict__ src, float* __restrict__ dst) {
  int r = blockIdx.x;
  int t = threadIdx.x;
  float acc = 0.f;
  #pragma unroll
  for (int i = 0; i < NS; ++i)
    acc += src[((long long)r * NS + i) * DIM + t];
  dst[(long long)r * DIM + t] = acc * (1.0f / NS);
}

// One wave per 16x16 output tile:  OUT[tm*16.., colOff + tn*16..] = act(A[MxK] @ W[Kx128])
// Uses V_WMMA_F32_16X16X4_F32 (fp32 A/B/C) — exact fp32 path, K stepped by 4.
// A 16x4 frag: lane half h holds K = k0+2h, k0+2h+1 for row M = lane%16 (contiguous pair).
// B 4x16 frag mirrors A: lane half h holds K = k0+2h, k0+2h+1 for col N = lane%16.
// C/D: VGPR r, lanes 0-15 -> M=r, N=lane; lanes 16-31 -> M=r+8, N=lane-16.
template <int K, bool RELU>
__global__ void k_gemm_wmma(const float* __restrict__ A,   // M x K
                            const float* __restrict__ W,   // K x 128
                            float* __restrict__ OUT,       // M x 256 (concat)
                            int colOff) {
  const int ntn = FEAT / 16;            // 8 N-tiles
  int tile = blockIdx.x;
  int tm = tile / ntn;
  int tn = tile % ntn;
  int lane = threadIdx.x;               // 0..31 (wave32, EXEC all ones)
  int half = lane >> 4;                 // 0 or 1
  int l16  = lane & 15;

  const float* Arow = A + (long long)(tm * 16 + l16) * K;  // A row m = l16
  const float* Wcol = W + tn * 16 + l16;                   // W col n = l16

  v8f c = {0.f, 0.f, 0.f, 0.f, 0.f, 0.f, 0.f, 0.f};
  #pragma unroll
  for (int k0 = 0; k0 < K; k0 += 4) {
    int ka = k0 + 2 * half;
    v2f a, b;
    a.x = Arow[ka];
    a.y = Arow[ka + 1];
    b.x = Wcol[(long long)ka * FEAT];
    b.y = Wcol[(long long)(ka + 1) * FEAT];
    // (neg_a, A, neg_b, B, c_mod, C, reuse_a, reuse_b)
    c = __builtin_amdgcn_wmma_f32_16x16x4_f32(false, a, false, b, (short)0, c,
                                              false, false);
  }

  long long rowBase = (long long)tm * 16 + 8 * half;
  int col = colOff + tn * 16 + l16;
  #pragma unroll
  for (int r = 0; r < 8; ++r) {
    float v = c[r];
    if (RELU) v = v > 0.f ? v : 0.f;
    OUT[(rowBase + r) * OUTD + col] = v;
  }
}

extern "C" void kernel_launch(void* const* d_in, const int* in_sizes, int n_in,
                              void* d_out, int out_size, void* d_ws, size_t ws_size,
                              hipStream_t stream) {
  (void)in_sizes; (void)n_in; (void)out_size; (void)ws_size;

  const float* features = (const float*)d_in[0];
  const int*   adj      = (const int*)d_in[1];
  const int*   batch1   = (const int*)d_in[2];
  const int*   cols1    = (const int*)d_in[3];
  const int*   cols2    = (const int*)d_in[4];
  const float* Ws0      = (const float*)d_in[5];
  const float* Wn0      = (const float*)d_in[6];
  const float* Ws1      = (const float*)d_in[7];
  const float* Wn1      = (const float*)d_in[8];
  float* out = (float*)d_out;

  // carve scratch from d_ws (~11.6 MB total)
  char* ws = (char*)d_ws;
  size_t off = 0;
  auto alloc = [&](size_t bytes) -> void* {
    void* p = ws + off;
    off += (bytes + 255) & ~(size_t)255;
    return p;
  };
  int*   n1  = (int*)  alloc((size_t)N1 * sizeof(int));
  float* X0  = (float*)alloc((size_t)BATCH * FEAT * sizeof(float));   // feat[batch1]
  float* X1  = (float*)alloc((size_t)N1 * FEAT * sizeof(float));      // feat[n1]
  float* M2  = (float*)alloc((size_t)N1 * FEAT * sizeof(float));      // mean25 of feat[n2]
  float* M1  = (float*)alloc((size_t)BATCH * FEAT * sizeof(float));   // mean10 of X1
  float* O0  = (float*)alloc((size_t)BATCH * OUTD * sizeof(float));   // layer0 hop0
  float* O1  = (float*)alloc((size_t)N1 * OUTD * sizeof(float));      // layer0 hop1
  float* MO1 = (float*)alloc((size_t)BATCH * OUTD * sizeof(float));   // mean10 of O1

  // --- sampling + gathers + means ---
  k_compute_n1<<<(N1 + 255) / 256, 256, 0, stream>>>(adj, batch1, cols1, n1);
  k_gather<<<BATCH, FEAT, 0, stream>>>(features, batch1, X0);
  k_gather<<<N1, FEAT, 0, stream>>>(features, n1, X1);
  k_mean25<<<N1, FEAT, 0, stream>>>(features, adj, n1, cols2, M2);
  k_mean_rows<FEAT, S1><<<BATCH, FEAT, 0, stream>>>(X1, M1);

  // --- layer 0: relu(concat([h @ Ws0, mean @ Wn0])) ---
  k_gemm_wmma<FEAT, true><<<(BATCH / 16) * 8, 32, 0, stream>>>(X0, Ws0, O0, 0);
  k_gemm_wmma<FEAT, true><<<(BATCH / 16) * 8, 32, 0, stream>>>(M1, Wn0, O0, FEAT);
  k_gemm_wmma<FEAT, true><<<(N1 / 16) * 8, 32, 0, stream>>>(X1, Ws0, O1, 0);
  k_gemm_wmma<FEAT, true><<<(N1 / 16) * 8, 32, 0, stream>>>(M2, Wn0, O1, FEAT);

  // --- layer 1: concat([out0 @ Ws1, mean10(out1) @ Wn1]) (no activation) ---
  k_mean_rows<OUTD, S1><<<BATCH, OUTD, 0, stream>>>(O1, MO1);
  k_gemm_wmma<OUTD, false><<<(BATCH / 16) * 8, 32, 0, stream>>>(O0, Ws1, out, 0);
  k_gemm_wmma<OUTD, false><<<(BATCH / 16) * 8, 32, 0, stream>>>(MO1, Wn1, out, FEAT);
}